// DecoderRNN_67662914781939
// MI455X (gfx1250) — compile-verified
//
#include <hip/hip_runtime.h>
#include <math.h>

// ---------------- CDNA5 WMMA types ----------------
typedef float v2f __attribute__((ext_vector_type(2)));
typedef float v8f __attribute__((ext_vector_type(8)));

#define HDIM   256
#define G4     1024          // 4*H
#define NLAYER 9
#define NWAVES 8             // waves per gate workgroup (256 threads, wave32)

// One GEMV pass: this wave's 8 K-chunks (K=4 each) of   gate += vec @ W^T.
// A (16x4 f32): row 0 = vec[k0..k0+3]; rows 1..15 zero (masked via cndmask).
// B (4x16 f32): B[k][n] = W[tile*16+n][k0+k]; one 8B global load per lane.
// Two accumulators break the WMMA->WMMA RAW dependency chain.
__device__ __forceinline__ void gemv_pass(const float* __restrict__ wrow,
                                          const float* vecp,
                                          int k0base, int koff, bool live,
                                          v8f& acc0, v8f& acc1)
{
    #pragma unroll
    for (int i = 0; i < 8; ++i) {
        const int k0 = k0base + 32 * i;          // this wave's chunk i
        // Unconditional LDS read (address valid for all lanes), then mask.
        const v2f av = *(const v2f*)(vecp + k0 + koff);
        v2f a;
        a.x = live ? av.x : 0.0f;                // v_cndmask, no branches
        a.y = live ? av.y : 0.0f;
        const v2f b = *(const v2f*)(wrow + k0 + koff);
        if (i & 1)
            acc1 = __builtin_amdgcn_wmma_f32_16x16x4_f32(
                false, a, false, b, (short)0, acc1, false, false);
        else
            acc0 = __builtin_amdgcn_wmma_f32_16x16x4_f32(
                false, a, false, b, (short)0, acc0, false, false);
    }
}

// ------------------------------------------------------------------
// Gate kernel: one workgroup per 16-gate output tile (64 WGs),
// 8 waves split the K=256(+256) reduction; partials combined via LDS.
// ------------------------------------------------------------------
__launch_bounds__(NWAVES * 32)
__global__ void lstm_gates_kernel(int layer,
                                  const float* __restrict__ x,      // [2]
                                  const float* __restrict__ h0,     // [9,256]
                                  const float* __restrict__ W_ih1,  // [1024,2]
                                  const float* __restrict__ W_ih,   // [8,1024,256]
                                  const float* __restrict__ W_hh,   // [9,1024,256]
                                  const float* __restrict__ b_ih,   // [9,1024]
                                  const float* __restrict__ b_hh,   // [9,1024]
                                  const float* __restrict__ ws_inp, // [256]
                                  float* __restrict__ ws_gates)     // [1024]
{
    __shared__ float s_hvec[HDIM];
    __shared__ float s_inp[HDIM];
    __shared__ float s_part[NWAVES * 16];

    const int tid  = threadIdx.x;            // 0..255
    const int lane = tid & 31;
    const int l16  = lane & 15;
    const bool lo  = lane < 16;
    // Wave index as an SGPR so loop bounds/addresses stay scalar.
    const int wave = __builtin_amdgcn_readfirstlane(tid >> 5);   // 0..7
    const int tile = blockIdx.x;             // 0..63
    const int row  = tile * 16 + l16;        // gate row for this lane's B column

    // Stage input vectors into LDS (one global read of each).
    s_hvec[tid] = h0[layer * HDIM + tid];
    if (layer > 0) s_inp[tid] = ws_inp[tid];
    __syncthreads();

    const float* Wh_row = W_hh + (size_t)layer * G4 * HDIM + (size_t)row * HDIM;

    const int  koff   = lo ? 0 : 2;          // B/A K offset for this lane half
    const bool live   = (l16 == 0);          // lanes 0 and 16 carry A row 0
    const int  k0base = wave * 4;            // chunk stride across waves = 32

    v8f acc0 = {0.f, 0.f, 0.f, 0.f, 0.f, 0.f, 0.f, 0.f};
    v8f acc1 = acc0;

    if (layer > 0) {
        const float* Wi_row =
            W_ih + (size_t)(layer - 1) * G4 * HDIM + (size_t)row * HDIM;
        // Warm up pass-1 weights while pass 0 runs (speculative prefetch).
        __builtin_prefetch(Wi_row + k0base, 0, 1);
        gemv_pass(Wh_row, s_hvec, k0base, koff, live, acc0, acc1);
        gemv_pass(Wi_row, s_inp,  k0base, koff, live, acc0, acc1);
    } else {
        gemv_pass(Wh_row, s_hvec, k0base, koff, live, acc0, acc1);
    }

    // D[0][n] lives in VGPR0 (acc[0]) of lanes 0..15.
    if (lo) s_part[wave * 16 + l16] = acc0[0] + acc1[0];
    __syncthreads();

    if (tid < 16) {
        const int r = tile * 16 + tid;
        float g = b_ih[layer * G4 + r] + b_hh[layer * G4 + r];
        #pragma unroll
        for (int w = 0; w < NWAVES; ++w) g += s_part[w * 16 + tid];
        if (layer == 0)
            g += x[0] * W_ih1[2 * r + 0] + x[1] * W_ih1[2 * r + 1];
        ws_gates[r] = g;
    }
}

// ------------------------------------------------------------------
// Pointwise LSTM cell update + residual-chain bookkeeping.
// One WG of 256 threads; also does the final W_out projection at k=8.
// ------------------------------------------------------------------
__launch_bounds__(HDIM)
__global__ void lstm_update_kernel(int layer,
                                   const float* __restrict__ c0,       // [9,256]
                                   const float* __restrict__ ws_gates, // [1024]
                                   float* __restrict__ ws_inp,         // [256]
                                   float* __restrict__ ws_hprev,       // [256]
                                   const float* __restrict__ W_out,    // [256]
                                   const float* __restrict__ b_out,    // [1]
                                   float* __restrict__ out)            // [1+2304+2304]
{
    __shared__ float s_h[HDIM];
    const int u = threadIdx.x; // 0..255

    const float gi = ws_gates[0 * HDIM + u];
    const float gf = ws_gates[1 * HDIM + u];
    const float gg = ws_gates[2 * HDIM + u];
    const float go = ws_gates[3 * HDIM + u];
    const float cc = c0[layer * HDIM + u];

    const float si = 1.0f / (1.0f + __expf(-gi));
    const float sf = 1.0f / (1.0f + __expf(-gf));
    const float so = 1.0f / (1.0f + __expf(-go));
    const float cn = sf * cc + si * tanhf(gg);
    const float hn = so * tanhf(cn);

    // Outputs: [0]=scalar, [1..2304]=hs, [2305..4608]=cs
    out[1 + layer * HDIM + u]                 = hn;
    out[1 + NLAYER * HDIM + layer * HDIM + u] = cn;

    // Residual chain: inp_{k+1} = h_k + h_{k-1}  (just h_0 after layer 0)
    const float hprev_old = (layer > 0) ? ws_hprev[u] : 0.0f;
    ws_inp[u]   = hn + hprev_old;
    ws_hprev[u] = hn;

    if (layer == NLAYER - 1) {
        s_h[u] = hn;
        __syncthreads();
        if (u == 0) {
            float acc = b_out[0];
            for (int i = 0; i < HDIM; ++i) acc += s_h[i] * W_out[i];
            out[0] = acc;
        }
    }
}

// ------------------------------------------------------------------
extern "C" void kernel_launch(void* const* d_in, const int* in_sizes, int n_in,
                              void* d_out, int out_size, void* d_ws, size_t ws_size,
                              hipStream_t stream) {
    const float* x     = (const float*)d_in[0];
    const float* h0    = (const float*)d_in[1];
    const float* c0    = (const float*)d_in[2];
    const float* W_ih1 = (const float*)d_in[3];
    const float* W_ih  = (const float*)d_in[4];
    const float* W_hh  = (const float*)d_in[5];
    const float* b_ih  = (const float*)d_in[6];
    const float* b_hh  = (const float*)d_in[7];
    const float* W_out = (const float*)d_in[8];
    const float* b_out = (const float*)d_in[9];
    float* out = (float*)d_out;

    float* ws       = (float*)d_ws;
    float* ws_gates = ws;            // 1024 floats
    float* ws_inp   = ws + G4;       // 256 floats
    float* ws_hprev = ws + G4 + 256; // 256 floats

    for (int k = 0; k < NLAYER; ++k) {
        lstm_gates_kernel<<<dim3(64), dim3(NWAVES * 32), 0, stream>>>(
            k, x, h0, W_ih1, W_ih, W_hh, b_ih, b_hh, ws_inp, ws_gates);
        lstm_update_kernel<<<dim3(1), dim3(HDIM), 0, stream>>>(
            k, c0, ws_gates, ws_inp, ws_hprev, W_out, b_out, out);
    }
}